// TurboQuantMSE_81604378624045
// MI455X (gfx1250) — compile-verified
//
#include <hip/hip_runtime.h>
#include <hip/hip_bf16.h>

typedef float v2f __attribute__((ext_vector_type(2)));
typedef float v8f __attribute__((ext_vector_type(8)));
typedef float f4  __attribute__((ext_vector_type(4)));
typedef int   i4  __attribute__((ext_vector_type(4)));

#define DD     1024   // transform dimension
#define MSTR   20     // padded word stride for 64x16 row-major staging (bank-conflict free, 16B aligned)
#define OTSTR  68     // padded word stride for 16x64 transposed staging
#define WBUF   1280   // words per staging buffer (64*20)
#define NWAVE  4
#define BLOCK  128

__device__ __forceinline__ float hsign(int r, int c) {
  return (__builtin_popcount(r & c) & 1) ? -1.0f : 1.0f;
}

// H4 (Hadamard-4) applied across VGPR groups {0..3},{4..7} of a WMMA f32 accumulator
__device__ __forceinline__ v8f h4mix(v8f d) {
  v8f r;
#pragma unroll
  for (int g = 0; g < 8; g += 4) {
    float s0 = d[g+0] + d[g+1], s1 = d[g+0] - d[g+1];
    float s2 = d[g+2] + d[g+3], s3 = d[g+2] - d[g+3];
    r[g+0] = s0 + s2; r[g+1] = s1 + s3; r[g+2] = s0 - s2; r[g+3] = s1 - s3;
  }
  return r;
}

// Full 1024-pt FWHT for one row held in Ab (64x16, stride MSTR).
// Step A: along j = d[3:0] via WMMA f32 16x16x4, result stored transposed into Bb (16x64, stride OTSTR).
// Step B: along i[3:0] via WMMA + i[5:4] via in-register H4. Results -> yr[4] (v8f each).
__device__ __forceinline__ void fwht_wave(const float* __restrict__ Ab, float* __restrict__ Bb,
                                          const v2f hf[4], int c16, int half, v8f yr[4]) {
#pragma unroll
  for (int t = 0; t < 4; ++t) {
    v8f c = {};
#pragma unroll
    for (int k = 0; k < 4; ++k) {
      // A-fragment (16x4 f32): lanes 0-15 hold K=0,1 ; lanes 16-31 hold K=2,3
      v2f a = *(const v2f*)(Ab + (16*t + c16)*MSTR + 4*k + 2*half);
      c = __builtin_amdgcn_wmma_f32_16x16x4_f32(false, a, false, hf[k], (short)0, c, false, false);
    }
    // transposed store: lane (c16,half) holds col c16, rows 8*half..8*half+7 of tile t
    int w = c16*OTSTR + 16*t + 8*half;
    *(f4*)(Bb + w)     = (f4){c[0], c[1], c[2], c[3]};
    *(f4*)(Bb + w + 4) = (f4){c[4], c[5], c[6], c[7]};
  }
#pragma unroll
  for (int t = 0; t < 4; ++t) {
    v8f c = {};
    int jrow = 4*t + (c16 >> 2);
    int hi2  = c16 & 3;
#pragma unroll
    for (int k = 0; k < 4; ++k) {
      v2f a = *(const v2f*)(Bb + jrow*OTSTR + hi2*16 + 4*k + 2*half);
      c = __builtin_amdgcn_wmma_f32_16x16x4_f32(false, a, false, hf[k], (short)0, c, false, false);
    }
    yr[t] = h4mix(c);
  }
}

__global__ void __launch_bounds__(BLOCK)
tq_kernel(const float* __restrict__ x, const float* __restrict__ sigma,
          const float* __restrict__ centroids, const float* __restrict__ boundaries,
          float* __restrict__ xhat, int* __restrict__ idxout, int nrow)
{
  extern __shared__ float smem[];
  const int lane = threadIdx.x & 31;
  const int wave = threadIdx.x >> 5;
  const int half = lane >> 4;
  const int c16  = lane & 15;

  float* Ab = smem + wave * (2 * WBUF);
  float* Bb = Ab + WBUF;
  int*   Bi = (int*)Bb;
  float* ct = smem + NWAVE * 2 * WBUF;   // 16-entry centroid table (bank-broadcast friendly)

  if (threadIdx.x < 16) ct[threadIdx.x] = centroids[threadIdx.x];
  __syncthreads();

  // H16 B-fragments (4x16 slices): lane col = c16; vgpr0/1 = rows 4k+2*half, +1
  v2f hf[4];
#pragma unroll
  for (int k = 0; k < 4; ++k) {
    int r0 = 4*k + 2*half;
    hf[k][0] = hsign(r0,     c16);
    hf[k][1] = hsign(r0 + 1, c16);
  }

  // boundaries pre-scaled by 32: compare unnormalized FWHT output directly
  float bs[15];
#pragma unroll
  for (int k = 0; k < 15; ++k) bs[k] = 32.0f * boundaries[k];

  const int rstride = NWAVE * gridDim.x;
  for (int row = blockIdx.x * NWAVE + wave; row < nrow; row += rstride) {
    const float* xr = x + (size_t)row * DD;

    // ---- load x*sigma into Ab (64x16 padded row-major), coalesced b128 ----
#pragma unroll
    for (int it = 0; it < 8; ++it) {
      int d0 = it*128 + lane*4;
      f4 xv = *(const f4*)(xr + d0);
      f4 sg = *(const f4*)(sigma + d0);
      *(f4*)(Ab + (d0 >> 4)*MSTR + (d0 & 15)) = xv * sg;
      if (row + rstride < nrow)
        __builtin_prefetch(xr + (size_t)rstride * DD + d0, 0, 1);
    }

    // ===== FWHT #1 =====
    v8f yr[4];
    fwht_wave(Ab, Bb, hf, c16, half, yr);

    // ---- quantize (exact binary-search tree in registers), scatter yhat->Ab, idx->Bi ----
#pragma unroll
    for (int t = 0; t < 4; ++t) {
#pragma unroll
      for (int v = 0; v < 8; ++v) {
        float acc = yr[t][v];
        bool  c1 = acc > bs[7];
        float l2 = c1 ? bs[11] : bs[3];
        bool  c2 = acc > l2;
        float l3 = c1 ? (c2 ? bs[13] : bs[9]) : (c2 ? bs[5] : bs[1]);
        bool  c3 = acc > l3;
        float e0 = c3 ? bs[2]  : bs[0];
        float e1 = c3 ? bs[6]  : bs[4];
        float e2 = c3 ? bs[10] : bs[8];
        float e3 = c3 ? bs[14] : bs[12];
        float l4 = c1 ? (c2 ? e3 : e2) : (c2 ? e1 : e0);
        bool  c4 = acc > l4;
        int idx = ((int)c1 << 3) | ((int)c2 << 2) | ((int)c3 << 1) | (int)c4;
        float yh = ct[idx];
        // element position: d = hi2'*256 + lo4'*16 + j'
        int h = v & 3;
        int q = (v >> 2) + 2*half;
        int w = (h*16 + c16)*MSTR + 4*t + q;   // M-layout word of index d
        Ab[w] = yh;
        Bi[w] = idx;
      }
    }

    // ---- store indices (coalesced int4) ----
    int* ir = idxout + (size_t)row * DD;
#pragma unroll
    for (int it = 0; it < 8; ++it) {
      int d0 = it*128 + lane*4;
      i4 vi = *(const i4*)(Bi + (d0 >> 4)*MSTR + (d0 & 15));
      *(i4*)(ir + d0) = vi;
    }

    // ===== FWHT #2 (inverse, on y_hat in Ab) =====
    fwht_wave(Ab, Bb, hf, c16, half, yr);

    // scatter transform result back into Ab (M layout)
#pragma unroll
    for (int t = 0; t < 4; ++t) {
#pragma unroll
      for (int v = 0; v < 8; ++v) {
        int h = v & 3;
        int q = (v >> 2) + 2*half;
        Ab[(h*16 + c16)*MSTR + 4*t + q] = yr[t][v];
      }
    }

    // ---- final: x_hat = sigma * fwht(yhat) / 32, coalesced float4 ----
    float* orow = xhat + (size_t)row * DD;
#pragma unroll
    for (int it = 0; it < 8; ++it) {
      int d0 = it*128 + lane*4;
      f4 vv = *(const f4*)(Ab + (d0 >> 4)*MSTR + (d0 & 15));
      f4 sg = *(const f4*)(sigma + d0);
      *(f4*)(orow + d0) = vv * sg * 0.03125f;
    }
  }
}

extern "C" void kernel_launch(void* const* d_in, const int* in_sizes, int n_in,
                              void* d_out, int out_size, void* d_ws, size_t ws_size,
                              hipStream_t stream) {
  const float* x     = (const float*)d_in[0];
  const float* sigma = (const float*)d_in[1];
  const float* cent  = (const float*)d_in[2];
  const float* bnd   = (const float*)d_in[3];
  int nrow = in_sizes[0] / DD;

  float* xhat = (float*)d_out;
  int*   idxo = (int*)d_out + (size_t)nrow * DD;

  int grid = 2048;  // grid-stride over rows: 2048 blocks * 4 waves = 8192 row-workers
  size_t shmem = (size_t)(NWAVE * 2 * WBUF + 16) * sizeof(float);  // ~41 KB
  hipLaunchKernelGGL(tq_kernel, dim3(grid), dim3(BLOCK), shmem, stream,
                     x, sigma, cent, bnd, xhat, idxo, nrow);
}